// MemoryGraph_79980880986537
// MI455X (gfx1250) — compile-verified
//
#include <hip/hip_runtime.h>

// MemoryGraph recurrent cell network for MI455X (gfx1250).
// One workgroup per (b, cell): state (h, W, hebbian) stays in LDS across all
// T=16 steps; all matmuls run on v_wmma_f32_16x16x4_f32 tiles.

#define BQ   8
#define TQ   16
#define NCQ  128
#define NcQ  64
#define DnQ  64
#define AQ   4
#define HmQ  128
#define GAMMA_MAX 0.5f

#define LDA 66    // padded row stride (floats) for 64-wide LDS tiles
#define LDH 130   // padded row stride (floats) for 128-wide LDS tile

typedef float v2f __attribute__((ext_vector_type(2)));
typedef float v8f __attribute__((ext_vector_type(8)));

__device__ __forceinline__ v8f wmma_f32_4(v2f a, v2f b, v8f c) {
  return __builtin_amdgcn_wmma_f32_16x16x4_f32(false, a, false, b, (short)0, c,
                                               false, false);
}

__device__ __forceinline__ float fast_tanh(float v) {
#if __has_builtin(__builtin_amdgcn_tanhf)
  return __builtin_amdgcn_tanhf(v);        // V_TANH_F32 (trans, co-executes with WMMA)
#elif __has_builtin(__builtin_amdgcn_tanh_f32)
  return __builtin_amdgcn_tanh_f32(v);
#else
  return tanhf(v);
#endif
}

__launch_bounds__(256, 4)                  // cap at 256 VGPRs: no vgpr-msb toggles
__global__ void memgraph_kernel(
    const float* __restrict__ x,            // [B,T,NC,Dn]
    const float* __restrict__ h0,           // [B,NC,Nc,Dn]
    const float* __restrict__ W0,           // [B,NC,Nc,Nc]
    const float* __restrict__ heb0,         // [B,NC,Nc,Nc]
    const float* __restrict__ neuron_id,    // [NC,Nc,Dn]
    const float* __restrict__ msg_w1,       // [Hm,Dn]
    const float* __restrict__ msg_b1,       // [Hm]
    const float* __restrict__ msg_w2,       // [Dn,Hm]
    const float* __restrict__ msg_b2,       // [Dn]
    const float* __restrict__ inject_w,     // [NC,A*Dn,Dn]
    const float* __restrict__ inject_b,     // [NC,A*Dn]
    const float* __restrict__ W_decay_logit,     // [NC,Nc]
    const float* __restrict__ decay_gamma_logit, // [NC,Nc]
    const float* __restrict__ heb_decay_logit,   // [NC,Nc]
    const int*   __restrict__ in_port,      // [NC,A]
    const int*   __restrict__ out_port,     // [NC,A]
    float* __restrict__ out)                // [B,T,NC*Dn]
{
  __shared__ float sW  [NcQ][LDA];   // slow weights (decayed in place)
  __shared__ float sHeb[NcQ][LDA];   // hebbian fast weights
  __shared__ float sH  [NcQ][LDA];   // h  ->  m  ->  h_new (phase-rotated)
  __shared__ float sHin[NcQ][LDA];   // h_in
  __shared__ float sHid[NcQ][LDH];   // phase 2-3: (W+heb) overlay; phase 4-5: MLP hidden
  __shared__ float sXt[DnQ];
  __shared__ float sGamma[NcQ], sHg[NcQ], sWg[NcQ];
  __shared__ int   sIp[AQ], sOp[AQ];

  float* const sWH = &sHid[0][0];    // [NcQ x LDA] overlay, flat-indexed

  const int tid  = threadIdx.x;
  const int lane = tid & 31;
  const int wv   = tid >> 5;          // wave id 0..7
  const int half = lane >> 4;         // 0/1
  const int ln   = lane & 15;         // tile row (A) / tile col (B,C,D)

  const int bid = blockIdx.x;
  const int c   = bid & (NCQ - 1);
  const int b   = bid >> 7;

  // ---------------- load initial state + per-cell constants ----------------
  {
    const float* W0p  = W0   + (size_t)(b * NCQ + c) * NcQ * NcQ;
    const float* Hb0p = heb0 + (size_t)(b * NCQ + c) * NcQ * NcQ;
    const float* H0p  = h0   + (size_t)(b * NCQ + c) * NcQ * DnQ;
    for (int i = tid; i < NcQ * NcQ; i += 256) {
      int r = i >> 6, q = i & 63;
      sW[r][q]   = W0p[i];
      sHeb[r][q] = Hb0p[i];
      sH[r][q]   = H0p[i];
    }
    if (tid < NcQ) {
      sGamma[tid] = GAMMA_MAX / (1.f + __expf(-decay_gamma_logit[c * NcQ + tid]));
      sHg[tid]    = GAMMA_MAX / (1.f + __expf(-heb_decay_logit[c * NcQ + tid]));
      sWg[tid]    = GAMMA_MAX / (1.f + __expf(-W_decay_logit[c * NcQ + tid]));
    }
    if (tid < AQ) {
      sIp[tid] = in_port[c * AQ + tid];
      sOp[tid] = out_port[c * AQ + tid];
    }
  }
  __syncthreads();

  for (int t = 0; t < TQ; ++t) {
    // -------- phase 1: h_in := h ; stage x_t --------
    for (int i = tid; i < NcQ * NcQ; i += 256) {
      int r = i >> 6, q = i & 63;
      sHin[r][q] = sH[r][q];
    }
    if (tid < DnQ)
      sXt[tid] = x[(((size_t)b * TQ + t) * NCQ + c) * DnQ + tid];
    __syncthreads();

    // -------- phase 2: injection matvec + scatter-add ; precompute W+heb --------
    {
      const float* wrow = inject_w + ((size_t)c * AQ * DnQ + tid) * DnQ;
      float acc = inject_b[c * AQ * DnQ + tid];
      #pragma unroll
      for (int k = 0; k < DnQ; k += 4) {
        acc += wrow[k] * sXt[k] + wrow[k + 1] * sXt[k + 1] +
               wrow[k + 2] * sXt[k + 2] + wrow[k + 3] * sXt[k + 3];
      }
      int a = tid >> 6, d = tid & 63;
      atomicAdd(&sHin[sIp[a]][d], acc);   // duplicate ports must accumulate
    }
    for (int i = tid; i < NcQ * NcQ; i += 256) {   // fast-weight sum, read 4x later
      int r = i >> 6, q = i & 63;
      sWH[r * LDA + q] = sW[r][q] + sHeb[r][q];
    }
    __syncthreads();

    // -------- phase 3: m = (W + heb) @ h_in  -> sH --------
    for (int tile = wv * 2; tile < wv * 2 + 2; ++tile) {
      int ti = tile >> 2, tj = tile & 3;
      v8f acc = {};
      const float* arow = sWH + (ti * 16 + ln) * LDA;
      #pragma unroll
      for (int kk = 0; kk < 16; ++kk) {
        int kc = kk * 4 + half * 2;
        v2f av; av.x = arow[kc];            av.y = arow[kc + 1];
        v2f bv; bv.x = sHin[kc][tj * 16 + ln];
                bv.y = sHin[kc + 1][tj * 16 + ln];
        acc = wmma_f32_4(av, bv, acc);
      }
      #pragma unroll
      for (int r = 0; r < 8; ++r)
        sH[ti * 16 + half * 8 + r][tj * 16 + ln] = acc[r];
    }
    __syncthreads();

    // -------- phase 4: hid = tanh(m @ w1^T + b1)  -> sHid (overlay now dead) --------
    for (int idx = wv * 4; idx < wv * 4 + 4; ++idx) {
      int ti = idx >> 3, tj = idx & 7;        // output [Nc=64, Hm=128]
      v8f acc = {};
      const float* w1r = msg_w1 + (size_t)(tj * 16 + ln) * DnQ;
      #pragma unroll
      for (int kk = 0; kk < 16; ++kk) {
        int kc = kk * 4 + half * 2;
        v2f av; av.x = sH[ti * 16 + ln][kc]; av.y = sH[ti * 16 + ln][kc + 1];
        v2f bv; bv.x = w1r[kc];              bv.y = w1r[kc + 1];
        acc = wmma_f32_4(av, bv, acc);
      }
      float bias = msg_b1[tj * 16 + ln];
      #pragma unroll
      for (int r = 0; r < 8; ++r)
        sHid[ti * 16 + half * 8 + r][tj * 16 + ln] = fast_tanh(acc[r] + bias);
    }
    __syncthreads();

    // -------- phase 5: m2 = hid @ w2^T + b2 ; fuse gated state update -> sH --------
    for (int tile = wv * 2; tile < wv * 2 + 2; ++tile) {
      int ti = tile >> 2, tj = tile & 3;
      v8f acc = {};
      const float* w2r = msg_w2 + (size_t)(tj * 16 + ln) * HmQ;
      #pragma unroll
      for (int kk = 0; kk < 32; ++kk) {       // K = Hm = 128
        int kc = kk * 4 + half * 2;
        v2f av; av.x = sHid[ti * 16 + ln][kc]; av.y = sHid[ti * 16 + ln][kc + 1];
        v2f bv; bv.x = w2r[kc];                bv.y = w2r[kc + 1];
        acc = wmma_f32_4(av, bv, acc);
      }
      float b2 = msg_b2[tj * 16 + ln];
      #pragma unroll
      for (int r = 0; r < 8; ++r) {
        int row = ti * 16 + half * 8 + r, col = tj * 16 + ln;
        float g   = sGamma[row];
        float nid = neuron_id[((size_t)c * NcQ + row) * DnQ + col];
        sH[row][col] = (1.f - g) * sHin[row][col] + g * fast_tanh(acc[r] + b2 + nid);
      }
    }
    __syncthreads();

    // -------- phase 6: hebbian outer product + W decay + readout --------
    for (int tile = wv * 2; tile < wv * 2 + 2; ++tile) {
      int ti = tile >> 2, tj = tile & 3;
      v8f acc = {};
      #pragma unroll
      for (int kk = 0; kk < 16; ++kk) {       // K = Dn = 64 (a @ a^T)
        int kc = kk * 4 + half * 2;
        v2f av; av.x = sH[ti * 16 + ln][kc]; av.y = sH[ti * 16 + ln][kc + 1];
        v2f bv; bv.x = sH[tj * 16 + ln][kc]; bv.y = sH[tj * 16 + ln][kc + 1];
        acc = wmma_f32_4(av, bv, acc);
      }
      const float s = 1.f / 64.f;             // (Dn^-0.5)^2 on both factors
      #pragma unroll
      for (int r = 0; r < 8; ++r) {
        int row = ti * 16 + half * 8 + r, col = tj * 16 + ln;
        float hg = sHg[row];
        float v  = (row == col) ? 0.f
                 : (1.f - hg) * sHeb[row][col] + hg * acc[r] * s;
        sHeb[row][col] = v;
      }
    }
    for (int i = tid; i < NcQ * NcQ; i += 256) {   // W *= (1 - wg) per row
      int r = i >> 6, q = i & 63;
      sW[r][q] *= (1.f - sWg[r]);
    }
    if (tid < DnQ) {                                // readout from output ports
      float s4 = sH[sOp[0]][tid] + sH[sOp[1]][tid] +
                 sH[sOp[2]][tid] + sH[sOp[3]][tid];
      // mean over A=4 then * A^-0.5  ==  * 0.125
      out[(((size_t)b * TQ + t) * NCQ + c) * DnQ + tid] = s4 * 0.125f;
    }
    __syncthreads();
  }
}

extern "C" void kernel_launch(void* const* d_in, const int* in_sizes, int n_in,
                              void* d_out, int out_size, void* d_ws, size_t ws_size,
                              hipStream_t stream) {
  (void)in_sizes; (void)n_in; (void)out_size; (void)d_ws; (void)ws_size;
  memgraph_kernel<<<dim3(BQ * NCQ), dim3(256), 0, stream>>>(
      (const float*)d_in[0],  (const float*)d_in[1],  (const float*)d_in[2],
      (const float*)d_in[3],  (const float*)d_in[4],  (const float*)d_in[5],
      (const float*)d_in[6],  (const float*)d_in[7],  (const float*)d_in[8],
      (const float*)d_in[9],  (const float*)d_in[10], (const float*)d_in[11],
      (const float*)d_in[12], (const float*)d_in[13],
      (const int*)d_in[14],   (const int*)d_in[15],
      (float*)d_out);
}